// Word2SentBlock_60206851555568
// MI455X (gfx1250) — compile-verified
//
#include <hip/hip_runtime.h>

// ---------------- problem constants (match reference) ----------------
constexpr int Bn = 8;      // batch
constexpr int Ln = 4096;   // sequence length
constexpr int Dd = 1024;   // hidden dim
constexpr int Sn = 128;    // sentences per passage
constexpr int Pn = 4032;   // passage tokens (L - Q)

constexpr int CHUNK  = 252;           // tokens per block (multiple of 4), 16 * 252 = 4032
constexpr int NCHUNK = Pn / CHUNK;    // 16
constexpr int WAVES  = 8;             // 8 waves * 16 cols = 128 cols per block

typedef __attribute__((ext_vector_type(2))) float v2f;
typedef __attribute__((ext_vector_type(8))) float v8f;

// ---------------- init: zero sums (d_out) and counts (d_ws) ----------------
__global__ void w2s_zero_kernel(float* __restrict__ out, int* __restrict__ cnt) {
  int idx = blockIdx.x * blockDim.x + threadIdx.x;
  if (idx < Bn * Sn * Dd) out[idx] = 0.0f;
  if (idx < Bn * Sn)      cnt[idx] = 0;
}

// ---------------- per-sentence token counts ----------------
__global__ void w2s_count_kernel(const int* __restrict__ s2s, int* __restrict__ cnt) {
  int i = blockIdx.x * blockDim.x + threadIdx.x;
  if (i >= Bn * Pn) return;
  int b = i / Pn;
  atomicAdd(&cnt[b * Sn + s2s[i]], 1);
}

// ---------------- main: WMMA f32 segment-sum ----------------
// D(16x16) += onehot(16x4 f32) x data(4x16 f32), exact f32 accumulation.
// A layout (32-bit A 16x4): lanes 0-15 -> M=0..15, {v0,v1} = K{0,1};
//                           lanes 16-31 -> M=0..15, {v0,v1} = K{2,3}.
// B layout (4x16): v0 lanes0-15 = row K0 (N=0..15), lanes16-31 = row K2;
//                  v1 lanes0-15 = row K1,           lanes16-31 = row K3.
// C/D layout: VGPR r, lanes 0-15 -> M=r, N=lane; lanes 16-31 -> M=r+8, N=lane-16.
__global__ void __launch_bounds__(WAVES * 32)
w2s_segsum_wmma_kernel(const float* __restrict__ words,
                       const int*   __restrict__ bounds,
                       const int*   __restrict__ s2s,
                       float*       __restrict__ out) {
  __shared__ alignas(16) int lds_ids[CHUNK];

  const int b    = blockIdx.z;
  const int p0   = blockIdx.x * CHUNK;
  const int lane = threadIdx.x & 31;
  const int w    = threadIdx.x >> 5;
  const int col0 = (blockIdx.y * WAVES + w) * 16;

  const int q = bounds[b * 2];                       // passage start (dynamic slice)
  const float* __restrict__ base = words + ((size_t)b * Ln + (size_t)q + p0) * Dd;
  const int*   __restrict__ ids_g = s2s + (size_t)b * Pn + p0;

  // stage this chunk's segment ids into LDS once per block
  for (int i = threadIdx.x; i < CHUNK; i += blockDim.x) lds_ids[i] = ids_g[i];
  __syncthreads();

  const int  hi  = (lane >= 16) ? 1 : 0;
  const int  m   = lane & 15;
  const int  col = col0 + m;
  const float* __restrict__ colp = base + col;       // &data[p0 + 0][col]

  v8f c = {0.f, 0.f, 0.f, 0.f, 0.f, 0.f, 0.f, 0.f};
  int sbase = lds_ids[0];                            // wave-uniform sentence window base

  for (int k = 0; k < CHUNK / 4; ++k) {
    const int4 ids = ((const int4*)lds_ids)[k];      // LDS broadcast: wave-uniform quad ids
    const int  r0  = 4 * k + (hi ? 2 : 0);

    v2f bm;                                          // B: rows (r0, r0+1) at this column
    bm.x = colp[(size_t)(r0)     * Dd];
    bm.y = colp[(size_t)(r0 + 1) * Dd];

    // prefetch ~16 tokens ahead (speculative; dropped silently if past mapping)
    if (((k & 3) == 0) && (p0 + r0 + 16 < Pn))
      __builtin_prefetch(&colp[(size_t)(r0 + 16) * Dd], 0, 1);

    const int i0 = hi ? ids.z : ids.x;               // per-half token ids for A columns
    const int i1 = hi ? ids.w : ids.y;

    for (;;) {                                       // wave-uniform window loop
#if __has_builtin(__builtin_amdgcn_wmma_f32_16x16x4_f32)
      v2f a;
      // one-hot: ids outside [sbase, sbase+16) match no row -> contribute zero
      a.x = (i0 - sbase == m) ? 1.0f : 0.0f;
      a.y = (i1 - sbase == m) ? 1.0f : 0.0f;
      c = __builtin_amdgcn_wmma_f32_16x16x4_f32(
              /*neg_a=*/false, a, /*neg_b=*/false, bm,
              /*c_mod=*/(short)0, c, /*reuse_a=*/false, /*reuse_b=*/false);
#else
      // Exact emulation of the WMMA above with identical operand/result layout.
      // ids are wave-uniform, so A[M,k] is recomputable per lane; the two B
      // rows held by the opposite half-wave come from one xor-shuffle.
      {
        float ox = __shfl_xor(bm.x, 16);
        float oy = __shfl_xor(bm.y, 16);
        float bk0 = hi ? ox   : bm.x;   // B[K=0, m]
        float bk1 = hi ? oy   : bm.y;   // B[K=1, m]
        float bk2 = hi ? bm.x : ox;     // B[K=2, m]
        float bk3 = hi ? bm.y : oy;     // B[K=3, m]
        #pragma unroll
        for (int r = 0; r < 8; ++r) {
          int Mr = r + (hi ? 8 : 0);
          float acc = c[r];
          if (ids.x - sbase == Mr) acc += bk0;
          if (ids.y - sbase == Mr) acc += bk1;
          if (ids.z - sbase == Mr) acc += bk2;
          if (ids.w - sbase == Mr) acc += bk3;
          c[r] = acc;
        }
      }
#endif
      if (ids.w < sbase + 16) break;                 // whole quad inside window -> done

      // flush accumulated window rows (skip exact-zero = unused sentence rows)
      #pragma unroll
      for (int r = 0; r < 8; ++r) {
        float v = c[r];
        if (v != 0.0f) {
          int s = sbase + r + (hi ? 8 : 0);
          atomicAdd(&out[((size_t)b * Sn + s) * Dd + col], v);
        }
      }
      c = (v8f){0.f, 0.f, 0.f, 0.f, 0.f, 0.f, 0.f, 0.f};
      // rebase to first id not covered by old window
      sbase = (ids.x >= sbase + 16) ? ids.x
            : (ids.y >= sbase + 16) ? ids.y
            : (ids.z >= sbase + 16) ? ids.z
            : ids.w;
    }
  }

  // final flush for this chunk/tile
  #pragma unroll
  for (int r = 0; r < 8; ++r) {
    float v = c[r];
    if (v != 0.0f) {
      int s = sbase + r + (hi ? 8 : 0);
      if (s < Sn) atomicAdd(&out[((size_t)b * Sn + s) * Dd + col], v);
    }
  }
}

// ---------------- finalize: divide sums by max(count, 1) ----------------
__global__ void w2s_divide_kernel(float* __restrict__ out, const int* __restrict__ cnt) {
  int idx = blockIdx.x * blockDim.x + threadIdx.x;
  if (idx >= Bn * Sn * Dd) return;
  float c = (float)cnt[idx / Dd];
  out[idx] = out[idx] / fmaxf(c, 1.0f);
}

// ---------------- launcher ----------------
extern "C" void kernel_launch(void* const* d_in, const int* in_sizes, int n_in,
                              void* d_out, int out_size, void* d_ws, size_t ws_size,
                              hipStream_t stream) {
  const float* words  = (const float*)d_in[0];   // [B, L, D] f32
  const int*   bounds = (const int*)  d_in[1];   // [B, 2]
  const int*   s2s    = (const int*)  d_in[2];   // [B, P]
  float*       out    = (float*)d_out;           // [B, S, D] f32
  int*         cnt    = (int*)d_ws;              // [B, S] counts (scratch)

  (void)in_sizes; (void)n_in; (void)out_size; (void)ws_size;

  const int NOUT = Bn * Sn * Dd;
  w2s_zero_kernel<<<(NOUT + 255) / 256, 256, 0, stream>>>(out, cnt);
  w2s_count_kernel<<<(Bn * Pn + 255) / 256, 256, 0, stream>>>(s2s, cnt);

  dim3 grid(NCHUNK, Dd / (WAVES * 16), Bn);      // 16 x 8 x 8 = 1024 blocks
  w2s_segsum_wmma_kernel<<<grid, WAVES * 32, 0, stream>>>(words, bounds, s2s, out);

  w2s_divide_kernel<<<(NOUT + 255) / 256, 256, 0, stream>>>(out, cnt);
}